// eSCNMDBackbone_31928786878749
// MI455X (gfx1250) — compile-verified
//
#include <hip/hip_runtime.h>
#include <math.h>

#define TB 128  // 4 wave32 waves; E=800000 is an exact multiple of 128

#if __has_builtin(__builtin_amdgcn_global_load_async_to_lds_b32)
#define USE_ASYNC_LDS 1
#else
#define USE_ASYNC_LDS 0
#endif

#if __has_builtin(__builtin_amdgcn_wmma_f32_16x16x4_f32)
#define USE_WMMA 1
#else
#define USE_WMMA 0
#endif

typedef __attribute__((address_space(1))) int as1_int;
typedef __attribute__((address_space(3))) int as3_int;
typedef __attribute__((ext_vector_type(4))) float v4f;   // for nontemporal vector stores
typedef __attribute__((ext_vector_type(2))) float v2f;   // packed f32 math / WMMA A,B operands
typedef __attribute__((ext_vector_type(8))) float v8f;   // WMMA C/D accumulator

__device__ __forceinline__ float fast_rsq(float x) {
#if __has_builtin(__builtin_amdgcn_rsqf)
    return __builtin_amdgcn_rsqf(x);   // v_rsq_f32
#else
    return 1.0f / sqrtf(x);
#endif
}

__global__ __launch_bounds__(TB) void wigner_fused_kernel(
    const float* __restrict__ edv,   // (E,3)
    const float* __restrict__ rnd,   // (E,3)
    const float* __restrict__ Jd1,   // (3,3)
    const float* __restrict__ Jd2,   // (5,5)
    const float* __restrict__ to_m,  // (9,9)
    float* __restrict__ outR,        // (E,3,3)
    float* __restrict__ outWm,       // (E,9,9)
    float* __restrict__ outWmi,      // (E,9,9)
    int E)
{
    __shared__ float sC[115];                        // [0..8]=Jd1, [9..33]=Jd2, [34..114]=to_m
    __shared__ __align__(16) float sR[TB * 9];       // R staging tile
    __shared__ __align__(16) float sWm[TB * 81];     // wm staging tile
    __shared__ __align__(16) float sWi[TB * 81];     // wm^T staging tile
#if USE_WMMA
    __shared__ __align__(16) float sWg[TB * 81];     // dense wigner tile (WMMA B source)
#endif

    const int t = threadIdx.x;

    // ---- Stage the small constant matrices into LDS (async DMA path) ----
    if (t < 115) {
        const float* g = (t < 9) ? (Jd1 + t)
                       : (t < 34) ? (Jd2 + (t - 9))
                                  : (to_m + (t - 34));
#if USE_ASYNC_LDS
        __builtin_amdgcn_global_load_async_to_lds_b32(
            (as1_int*)g, (as3_int*)(sC + t), 0, 0);
#else
        sC[t] = g[0];
#endif
    }
#if USE_ASYNC_LDS
#if __has_builtin(__builtin_amdgcn_s_wait_asynccnt)
    __builtin_amdgcn_s_wait_asynccnt(0);
#else
    asm volatile("s_wait_asynccnt 0" ::: "memory");
#endif
#endif
    __syncthreads();

    const float* sJ1 = sC;        // 3x3
    const float* sJ2 = sC + 9;    // 5x5
    const float* sT  = sC + 34;   // 9x9

    const int blockBase = blockIdx.x * TB;
    int nvalid = E - blockBase;
    if (nvalid > TB) nvalid = TB;
    const int n = blockBase + t;
    const int m = (n < E) ? n : (E - 1);   // clamp for loads; stores are range-guarded

    // ---------------- per-edge frame construction ----------------
    float vx = edv[3 * m + 0], vy = edv[3 * m + 1], vz = edv[3 * m + 2];
    float inv = fast_rsq(vx * vx + vy * vy + vz * vz);
    float nx0 = vx * inv, nx1 = vy * inv, nx2 = vz * inv;

    float ex = rnd[3 * m + 0] - 0.5f, ey = rnd[3 * m + 1] - 0.5f, ez = rnd[3 * m + 2] - 0.5f;
    inv = fast_rsq(ex * ex + ey * ey + ez * ez);
    ex *= inv; ey *= inv; ez *= inv;

    float bx = -ey, by = ex, bz = ez;       // e2b
    float cx = ex, cy = -ez, cz = ey;       // e2c
    float d0 = fabsf(ex * nx0 + ey * nx1 + ez * nx2);
    float d1 = fabsf(bx * nx0 + by * nx1 + bz * nx2);
    if (d0 > d1) { ex = bx; ey = by; ez = bz; d0 = d1; }
    float d2 = fabsf(cx * nx0 + cy * nx1 + cz * nx2);
    if (d0 > d2) { ex = cx; ey = cy; ez = cz; }

    // norm_z = normalize(cross(norm_x, e2))
    float zx = nx1 * ez - nx2 * ey;
    float zy = nx2 * ex - nx0 * ez;
    float zz = nx0 * ey - nx1 * ex;
    inv = fast_rsq(zx * zx + zy * zy + zz * zz);
    zx *= inv; zy *= inv; zz *= inv;
    // norm_y = normalize(cross(norm_x, norm_z))
    float yx = nx1 * zz - nx2 * zy;
    float yy = nx2 * zx - nx0 * zz;
    float yz = nx0 * zy - nx1 * zx;
    inv = fast_rsq(yx * yx + yy * yy + yz * yz);
    yx *= inv; yy *= inv; yz *= inv;

    // R rows: [norm_z; norm_x; norm_y]
    float Rl[9] = { zx, zy, zz, nx0, nx1, nx2, yx, yy, yz };

    // ---- Euler angle sines/cosines, fully algebraic (no libm trig) ----
    float denA = zy * zy + yy * yy;
    float rhA = fast_rsq(denA);
    float sA = zy * rhA, cA = yy * rhA;
    if (denA == 0.0f) { sA = 0.0f; cA = 1.0f; }
    float xb = fminf(1.0f, fmaxf(-1.0f, nx1));
    float cB = xb;
    float sB = sqrtf(fmaxf(0.0f, 1.0f - xb * xb));
    float Rg00 = cA * zx - sA * yx;
    float Rg02 = cA * zz - sA * yz;
    float denG = Rg02 * Rg02 + Rg00 * Rg00;
    float rhG = fast_rsq(denG);
    float sG = Rg02 * rhG, cG = Rg00 * rhG;
    if (denG == 0.0f) { sG = 0.0f; cG = 1.0f; }
    float s2A = 2.0f * sA * cA, c2A = cA * cA - sA * sA;
    float s2B = 2.0f * sB * cB, c2B = cB * cB - sB * sB;
    float s2G = 2.0f * sG * cG, c2G = cG * cG - sG * sG;

    // ---------------- Wigner D1 = Z1(a) J1 Z1(b) J1 Z1(g) ----------------
    float A1[3][3], B1[3][3], C1[3][3], D1[3][3];
#pragma unroll
    for (int j = 0; j < 3; ++j) {
        A1[0][j] =  cA * sJ1[0 + j] + sA * sJ1[6 + j];
        A1[1][j] =  sJ1[3 + j];
        A1[2][j] = -sA * sJ1[0 + j] + cA * sJ1[6 + j];
    }
#pragma unroll
    for (int i = 0; i < 3; ++i) {
        B1[i][0] = cB * A1[i][0] - sB * A1[i][2];
        B1[i][1] = A1[i][1];
        B1[i][2] = sB * A1[i][0] + cB * A1[i][2];
    }
#pragma unroll
    for (int i = 0; i < 3; ++i)
#pragma unroll
        for (int j = 0; j < 3; ++j)
            C1[i][j] = B1[i][0] * sJ1[j] + B1[i][1] * sJ1[3 + j] + B1[i][2] * sJ1[6 + j];
#pragma unroll
    for (int i = 0; i < 3; ++i) {
        D1[i][0] = cG * C1[i][0] - sG * C1[i][2];
        D1[i][1] = C1[i][1];
        D1[i][2] = sG * C1[i][0] + cG * C1[i][2];
    }

    // ---------------- Wigner D2 = Z2(a) J2 Z2(b) J2 Z2(g) ----------------
    float A2[5][5], B2[5][5], C2[5][5], D2[5][5];
#pragma unroll
    for (int j = 0; j < 5; ++j) {
        A2[0][j] =  c2A * sJ2[0 + j]  + s2A * sJ2[20 + j];
        A2[1][j] =  cA  * sJ2[5 + j]  + sA  * sJ2[15 + j];
        A2[2][j] =  sJ2[10 + j];
        A2[3][j] = -sA  * sJ2[5 + j]  + cA  * sJ2[15 + j];
        A2[4][j] = -s2A * sJ2[0 + j]  + c2A * sJ2[20 + j];
    }
#pragma unroll
    for (int i = 0; i < 5; ++i) {
        B2[i][0] = c2B * A2[i][0] - s2B * A2[i][4];
        B2[i][1] = cB  * A2[i][1] - sB  * A2[i][3];
        B2[i][2] = A2[i][2];
        B2[i][3] = sB  * A2[i][1] + cB  * A2[i][3];
        B2[i][4] = s2B * A2[i][0] + c2B * A2[i][4];
    }
    // C2 = B2 * J2 (5x5x5): packed-f32 across column pairs -> v_pk_fma_f32
#pragma unroll
    for (int i = 0; i < 5; ++i) {
        v2f a01; a01.x = 0.0f; a01.y = 0.0f;
        v2f a23; a23.x = 0.0f; a23.y = 0.0f;
        float a4 = 0.0f;
#pragma unroll
        for (int k = 0; k < 5; ++k) {
            float b = B2[i][k];
            v2f bv; bv.x = b; bv.y = b;
            v2f j01; j01.x = sJ2[5 * k + 0]; j01.y = sJ2[5 * k + 1];
            v2f j23; j23.x = sJ2[5 * k + 2]; j23.y = sJ2[5 * k + 3];
            a01 += bv * j01;
            a23 += bv * j23;
            a4  += b * sJ2[5 * k + 4];
        }
        C2[i][0] = a01.x; C2[i][1] = a01.y;
        C2[i][2] = a23.x; C2[i][3] = a23.y;
        C2[i][4] = a4;
    }
#pragma unroll
    for (int i = 0; i < 5; ++i) {
        D2[i][0] = c2G * C2[i][0] - s2G * C2[i][4];
        D2[i][1] = cG  * C2[i][1] - sG  * C2[i][3];
        D2[i][2] = C2[i][2];
        D2[i][3] = sG  * C2[i][1] + cG  * C2[i][3];
        D2[i][4] = s2G * C2[i][0] + c2G * C2[i][4];
    }

#if USE_WMMA
    // ================= WMMA path: wm = to_m x blockdiag(1,D1,D2) on the matrix pipe =================
    // Stage dense per-edge wigner (B source) + R tile.
    {
        const int base81 = t * 81;   // stride 81 = 17 mod 64 -> bank-conflict free
        float W[9][9];
#pragma unroll
        for (int r = 0; r < 9; ++r)
#pragma unroll
            for (int j = 0; j < 9; ++j) W[r][j] = 0.0f;
        W[0][0] = 1.0f;
#pragma unroll
        for (int i = 0; i < 3; ++i)
#pragma unroll
            for (int j = 0; j < 3; ++j) W[1 + i][1 + j] = D1[i][j];
#pragma unroll
        for (int i = 0; i < 5; ++i)
#pragma unroll
            for (int j = 0; j < 5; ++j) W[4 + i][4 + j] = D2[i][j];
#pragma unroll
        for (int r = 0; r < 9; ++r)
#pragma unroll
            for (int j = 0; j < 9; ++j) sWg[base81 + r * 9 + j] = W[r][j];
#pragma unroll
        for (int i = 0; i < 9; ++i) sR[t * 9 + i] = Rl[i];
    }
    __syncthreads();

    {
        const int lane  = t & 31;
        const int wslot = t >> 5;            // wave slot 0..3, each owns 32 edge slots
        const int m16   = lane & 15;
        const bool hiHalf = lane >= 16;
        const int edgeTileBase = (wslot * 32) * 81;   // float offset of this wave's first edge slot

        // A = to_m in WMMA A-layout (16x4 f32, 2 VGPRs), 3 K-chunks covering K=0..11 (k>8 zeroed)
        v2f Aq[3];
#pragma unroll
        for (int q = 0; q < 3; ++q) {
            int kx = 4 * q + (hiHalf ? 2 : 0);
            int ky = kx + 1;
            float ax = (m16 < 9 && kx < 9) ? sT[m16 * 9 + kx] : 0.0f;
            float ay = (m16 < 9 && ky < 9) ? sT[m16 * 9 + ky] : 0.0f;
            v2f a; a.x = ax; a.y = ay; Aq[q] = a;
        }

        // 18 chains x 16 columns cover 32 edges x 9 output columns
        for (int c = 0; c < 18; ++c) {
            int coln = c * 16 + m16;          // 0..287 = (edge-in-wave, j) flattened
            int eloc = coln / 9;
            int j    = coln - eloc * 9;
            int wb   = edgeTileBase + eloc * 81;

            v2f Bq[3];
#pragma unroll
            for (int q = 0; q < 3; ++q) {
                int k0 = 4 * q + (hiHalf ? 2 : 0);
                int k1 = k0 + 1;
                float bx = (k0 <= 8) ? sWg[wb + k0 * 9 + j] : 0.0f;
                float by = (k1 <= 8) ? sWg[wb + k1 * 9 + j] : 0.0f;
                v2f b; b.x = bx; b.y = by; Bq[q] = b;
            }

            v8f acc = {};
            acc = __builtin_amdgcn_wmma_f32_16x16x4_f32(false, Aq[0], false, Bq[0],
                                                        (short)0, acc, false, false);
            acc = __builtin_amdgcn_wmma_f32_16x16x4_f32(false, Aq[1], false, Bq[1],
                                                        (short)0, acc, false, false);
            acc = __builtin_amdgcn_wmma_f32_16x16x4_f32(false, Aq[2], false, Bq[2],
                                                        (short)0, acc, false, false);

            // D layout: VGPR r = rows (r | r+8); keep rows m<=8 only
            const int mBias = hiHalf ? 8 : 0;
#pragma unroll
            for (int r = 0; r < 8; ++r) {
                int mm = r + mBias;
                if (mm <= 8) {
                    float dv = acc[r];
                    sWm[wb + mm * 9 + j] = dv;   // wm
                    sWi[wb + j * 9 + mm] = dv;   // wm^T (= wm_inv)
                }
            }
        }
    }
    __syncthreads();
#else
    // ================= scalar fallback: wm on VALU =================
    v2f D2p01[5], D2p23[5];
    float D2c4[5];
#pragma unroll
    for (int k = 0; k < 5; ++k) {
        v2f p01; p01.x = D2[k][0]; p01.y = D2[k][1];
        v2f p23; p23.x = D2[k][2]; p23.y = D2[k][3];
        D2p01[k] = p01; D2p23[k] = p23; D2c4[k] = D2[k][4];
    }
    {
        const int base81 = t * 81;
#pragma unroll
        for (int r = 0; r < 9; ++r) {
            float t0 = sT[r * 9 + 0];
            float t1 = sT[r * 9 + 1], t2 = sT[r * 9 + 2], t3 = sT[r * 9 + 3];
            float tk[5] = { sT[r * 9 + 4], sT[r * 9 + 5], sT[r * 9 + 6],
                            sT[r * 9 + 7], sT[r * 9 + 8] };
            float row[9];
            row[0] = t0;  // D0 == 1
#pragma unroll
            for (int j = 0; j < 3; ++j)
                row[1 + j] = t1 * D1[0][j] + t2 * D1[1][j] + t3 * D1[2][j];
            v2f w45; w45.x = 0.0f; w45.y = 0.0f;
            v2f w67; w67.x = 0.0f; w67.y = 0.0f;
            float w8 = 0.0f;
#pragma unroll
            for (int k = 0; k < 5; ++k) {
                v2f bv; bv.x = tk[k]; bv.y = tk[k];
                w45 += bv * D2p01[k];
                w67 += bv * D2p23[k];
                w8  += tk[k] * D2c4[k];
            }
            row[4] = w45.x; row[5] = w45.y; row[6] = w67.x; row[7] = w67.y; row[8] = w8;
#pragma unroll
            for (int j = 0; j < 9; ++j) {
                sWm[base81 + r * 9 + j] = row[j];
                sWi[base81 + j * 9 + r] = row[j];
            }
        }
#pragma unroll
        for (int i = 0; i < 9; ++i) sR[t * 9 + i] = Rl[i];
    }
    __syncthreads();
#endif

    // ---------------- coalesced non-temporal streaming stores ----------------
    {
        float* dst = outR + (long)blockBase * 9;
        const int tot = nvalid * 9, tot4 = tot >> 2;
        const v4f* s4 = (const v4f*)sR;
        v4f* d4 = (v4f*)dst;
        for (int i = t; i < tot4; i += TB) __builtin_nontemporal_store(s4[i], d4 + i);
        for (int f = (tot4 << 2) + t; f < tot; f += TB)
            __builtin_nontemporal_store(sR[f], dst + f);
    }
    {
        float* dst = outWm + (long)blockBase * 81;
        const int tot = nvalid * 81, tot4 = tot >> 2;
        const v4f* s4 = (const v4f*)sWm;
        v4f* d4 = (v4f*)dst;
        for (int i = t; i < tot4; i += TB) __builtin_nontemporal_store(s4[i], d4 + i);
        for (int f = (tot4 << 2) + t; f < tot; f += TB)
            __builtin_nontemporal_store(sWm[f], dst + f);
    }
    {
        float* dst = outWmi + (long)blockBase * 81;
        const int tot = nvalid * 81, tot4 = tot >> 2;
        const v4f* s4 = (const v4f*)sWi;
        v4f* d4 = (v4f*)dst;
        for (int i = t; i < tot4; i += TB) __builtin_nontemporal_store(s4[i], d4 + i);
        for (int f = (tot4 << 2) + t; f < tot; f += TB)
            __builtin_nontemporal_store(sWi[f], dst + f);
    }
}

extern "C" void kernel_launch(void* const* d_in, const int* in_sizes, int n_in,
                              void* d_out, int out_size, void* d_ws, size_t ws_size,
                              hipStream_t stream) {
    const float* edv  = (const float*)d_in[0];
    const float* rnd  = (const float*)d_in[1];
    const float* Jd1  = (const float*)d_in[2];
    const float* Jd2  = (const float*)d_in[3];
    const float* to_m = (const float*)d_in[4];
    const int E = in_sizes[0] / 3;
    if (E <= 0) return;

    float* outR   = (float*)d_out;
    float* outWm  = outR + (size_t)E * 9;
    float* outWmi = outWm + (size_t)E * 81;

    const int grid = (E + TB - 1) / TB;
    hipLaunchKernelGGL(wigner_fused_kernel, dim3(grid), dim3(TB), 0, stream,
                       edv, rnd, Jd1, Jd2, to_m, outR, outWm, outWmi, E);
}